// AttentionProbe_15066745274720
// MI455X (gfx1250) — compile-verified
//
#include <hip/hip_runtime.h>
#include <hip/hip_bf16.h>

// ---------------------------------------------------------------------------
// AttentionProbe: logits = (X q)/sqrt(D); alpha = softmax_N(logits);
// pooled = alpha^T X; scores = pooled . head_w + head_b
// B=32, N=4096, D=1024, f32. 512MB streamed once (online softmax fusion).
// ---------------------------------------------------------------------------

#define BQ 32
#define NQ 4096
#define DQ 1024
#define CHUNKS 16
#define ROWS_PER_BLOCK (NQ / CHUNKS)            // 256
#define WAVES_PER_BLOCK 8
#define ROWS_PER_WAVE (ROWS_PER_BLOCK / WAVES_PER_BLOCK)  // 32

typedef __attribute__((ext_vector_type(2))) float v2f;
typedef __attribute__((ext_vector_type(4))) float v4f;
typedef __attribute__((ext_vector_type(8))) float v8f;

// ---------------------------------------------------------------------------
// Pass 1: fused logits + online-softmax pooled partials.
// grid = (CHUNKS, B); block = 256 (8 wave32). Each wave: 32 rows, row held in
// registers between the q-dot and the rank-1 p update (single HBM pass).
// ---------------------------------------------------------------------------
__global__ __launch_bounds__(256) void probe_pass1(
    const float* __restrict__ x_full,
    const unsigned char* __restrict__ mask,
    const float* __restrict__ q,
    float* __restrict__ logits_ws,   // [B][N] raw (masked) logits
    float* __restrict__ part_p,      // [B][CHUNKS][D] unnormalized pooled partial
    float* __restrict__ part_ms)     // [B][CHUNKS][2]  (M, S)
{
    const int chunk = blockIdx.x;
    const int b     = blockIdx.y;
    const int tid   = threadIdx.x;
    const int wave  = tid >> 5;
    const int lane  = tid & 31;

    // lane owns columns { j*128 + 4*lane + t : j in 0..7, t in 0..3 }
    v4f qv[8];
#pragma unroll
    for (int j = 0; j < 8; ++j)
        qv[j] = *(const v4f*)(q + j * 128 + 4 * lane);

    v4f p[8];
#pragma unroll
    for (int j = 0; j < 8; ++j) p[j] = (v4f){0.f, 0.f, 0.f, 0.f};

    float m_run = -__builtin_inff();
    float s_run = 0.0f;

    const int row0 = chunk * ROWS_PER_BLOCK + wave * ROWS_PER_WAVE;
    const float* __restrict__ xb = x_full + (size_t)b * NQ * DQ;
    const unsigned char* __restrict__ mb = mask + (size_t)b * NQ;

    for (int r = 0; r < ROWS_PER_WAVE; ++r) {
        const int n = row0 + r;
        const float* __restrict__ xr = xb + (size_t)n * DQ;

        v4f xv[8];
        float dp = 0.f;
#pragma unroll
        for (int j = 0; j < 8; ++j) {
            // stream-once data: non-temporal 512B-per-wave coalesced loads
            xv[j] = __builtin_nontemporal_load((const v4f*)(xr + j * 128 + 4 * lane));
            dp += xv[j].x * qv[j].x + xv[j].y * qv[j].y +
                  xv[j].z * qv[j].z + xv[j].w * qv[j].w;
        }
        // butterfly reduce: every lane ends with the full dot (wave-uniform)
#pragma unroll
        for (int off = 16; off > 0; off >>= 1)
            dp += __shfl_xor(dp, off, 32);

        float logit = dp * 0.03125f;               // 1/sqrt(1024)
        if (!mb[n]) logit = -__builtin_inff();
        if (lane == 0) logits_ws[(size_t)b * NQ + n] = logit;

        float m_new = fmaxf(m_run, logit);
        if (m_new == -__builtin_inff()) continue;  // everything masked so far

        float w = __expf(logit - m_new);
        if (m_new > m_run) {                       // wave-uniform rescale branch
            float scale = __expf(m_run - m_new);   // -inf old max -> 0
            s_run *= scale;
#pragma unroll
            for (int j = 0; j < 8; ++j) p[j] *= scale;
        }
        m_run = m_new;
        s_run += w;
#pragma unroll
        for (int j = 0; j < 8; ++j) {
            p[j].x = __builtin_fmaf(w, xv[j].x, p[j].x);
            p[j].y = __builtin_fmaf(w, xv[j].y, p[j].y);
            p[j].z = __builtin_fmaf(w, xv[j].z, p[j].z);
            p[j].w = __builtin_fmaf(w, xv[j].w, p[j].w);
        }
    }

    // ---- combine the 8 wave-local online-softmax states through LDS ----
    __shared__ float p_lds[WAVES_PER_BLOCK][DQ];     // 32 KB
    __shared__ float ms_lds[WAVES_PER_BLOCK][2];

#pragma unroll
    for (int j = 0; j < 8; ++j)
        *(v4f*)&p_lds[wave][j * 128 + 4 * lane] = p[j];
    if (lane == 0) { ms_lds[wave][0] = m_run; ms_lds[wave][1] = s_run; }
    __syncthreads();

    float Mb = -__builtin_inff();
    float mv[WAVES_PER_BLOCK], sv[WAVES_PER_BLOCK];
#pragma unroll
    for (int w = 0; w < WAVES_PER_BLOCK; ++w) {
        mv[w] = ms_lds[w][0];
        sv[w] = ms_lds[w][1];
        Mb = fmaxf(Mb, mv[w]);
    }
    float Sb = 0.f, fw[WAVES_PER_BLOCK];
#pragma unroll
    for (int w = 0; w < WAVES_PER_BLOCK; ++w) {
        float f = (Mb == -__builtin_inff()) ? 0.f : __expf(mv[w] - Mb);
        fw[w] = f;
        Sb += sv[w] * f;
    }
    v4f acc = (v4f){0.f, 0.f, 0.f, 0.f};
#pragma unroll
    for (int w = 0; w < WAVES_PER_BLOCK; ++w)
        acc += fw[w] * *(const v4f*)&p_lds[w][4 * tid];

    float* pp = part_p + ((size_t)b * CHUNKS + chunk) * DQ;
    *(v4f*)(pp + 4 * tid) = acc;
    if (tid == 0) {
        part_ms[((size_t)b * CHUNKS + chunk) * 2 + 0] = Mb;
        part_ms[((size_t)b * CHUNKS + chunk) * 2 + 1] = Sb;
    }
}

// ---------------------------------------------------------------------------
// Pass 2: merge chunk partials -> pooled[b][d]; emit alpha from saved logits.
// grid = B blocks of 256 threads. Tiny traffic (~2.5 MB).
// ---------------------------------------------------------------------------
__global__ __launch_bounds__(256) void probe_pass2(
    const float* __restrict__ part_p,
    const float* __restrict__ part_ms,
    const float* __restrict__ logits_ws,
    float* __restrict__ pooled_ws,   // [B][D]
    float* __restrict__ alpha_out)   // d_out + B
{
    const int b = blockIdx.x;
    const int tid = threadIdx.x;

    float M = -__builtin_inff();
    float mv[CHUNKS], sv[CHUNKS];
#pragma unroll
    for (int c = 0; c < CHUNKS; ++c) {
        mv[c] = part_ms[((size_t)b * CHUNKS + c) * 2 + 0];
        sv[c] = part_ms[((size_t)b * CHUNKS + c) * 2 + 1];
        M = fmaxf(M, mv[c]);
    }
    float S = 0.f, fc[CHUNKS];
#pragma unroll
    for (int c = 0; c < CHUNKS; ++c) {
        float f = (M == -__builtin_inff()) ? 0.f : __expf(mv[c] - M);
        fc[c] = f;
        S += sv[c] * f;
    }
    const float invS = (S > 0.f) ? (1.f / S) : 0.f;

    v4f acc = (v4f){0.f, 0.f, 0.f, 0.f};
#pragma unroll
    for (int c = 0; c < CHUNKS; ++c)
        acc += fc[c] * *(const v4f*)(part_p + ((size_t)b * CHUNKS + c) * DQ + 4 * tid);
    acc *= invS;
    *(v4f*)(pooled_ws + (size_t)b * DQ + 4 * tid) = acc;

    const float* lg = logits_ws + (size_t)b * NQ;
    float* al = alpha_out + (size_t)b * NQ;
    for (int n = tid; n < NQ; n += 256)
        al[n] = __expf(lg[n] - M) * invS;     // -inf logit -> exact 0
}

// ---------------------------------------------------------------------------
// Pass 3: scores = pooled(32x1024) . head_w + head_b via V_WMMA_F32_16X16X4_F32.
// 1 block, 16 wave32: wave = (mtile in {0,1}) x (k-chunk of 128), dual
// accumulators break the WMMA->WMMA RAW chain; LDS reduce over k-chunks.
// A layout (f32 16x4): lanes 0-15 hold K=0,1 in v[0],v[1]; lanes 16-31 K=2,3.
// B (4x16) = head_w[k] broadcast across the 16 N columns.
// ---------------------------------------------------------------------------
__global__ __launch_bounds__(512) void probe_pass3(
    const float* __restrict__ pooled_ws,
    const float* __restrict__ head_w,
    const float* __restrict__ head_b,
    float* __restrict__ scores)      // d_out[0..31]
{
    __shared__ float partial[8][BQ];

    const int tid   = threadIdx.x;
    const int wave  = tid >> 5;
    const int lane  = tid & 31;
    const int mtile = wave & 1;       // rows [mtile*16, mtile*16+16)
    const int kc    = wave >> 1;      // k range [kc*128, kc*128+128)
    const int m     = (lane & 15) + mtile * 16;
    const int kb    = (lane >> 4) * 2;   // 0 for lanes 0-15, 2 for 16-31

    const float* __restrict__ prow = pooled_ws + (size_t)m * DQ;

    v8f c0 = {}; v8f c1 = {};
#pragma unroll
    for (int k0 = kc * 128; k0 < kc * 128 + 128; k0 += 8) {
        v2f a0 = { prow[k0 + kb],     prow[k0 + kb + 1] };
        v2f b0 = { head_w[k0 + kb],   head_w[k0 + kb + 1] };
        v2f a1 = { prow[k0 + 4 + kb], prow[k0 + 4 + kb + 1] };
        v2f b1 = { head_w[k0 + 4 + kb], head_w[k0 + 4 + kb + 1] };
        c0 = __builtin_amdgcn_wmma_f32_16x16x4_f32(false, a0, false, b0,
                                                   (short)0, c0, false, false);
        c1 = __builtin_amdgcn_wmma_f32_16x16x4_f32(false, a1, false, b1,
                                                   (short)0, c1, false, false);
    }
    v8f cs = c0 + c1;

    // D layout: VGPR v, lanes 0-15 -> row v (col=lane), lanes 16-31 -> row v+8.
    // All 16 columns are identical (B columns replicated), take col 0 / col 16.
    if (lane == 0) {
#pragma unroll
        for (int v = 0; v < 8; ++v) partial[kc][mtile * 16 + v] = cs[v];
    }
    if (lane == 16) {
#pragma unroll
        for (int v = 0; v < 8; ++v) partial[kc][mtile * 16 + 8 + v] = cs[v];
    }
    __syncthreads();

    if (tid < BQ) {
        float s = head_b[0];
#pragma unroll
        for (int k = 0; k < 8; ++k) s += partial[k][tid];
        scores[tid] = s;
    }
}

// ---------------------------------------------------------------------------
// Launch
// ---------------------------------------------------------------------------
extern "C" void kernel_launch(void* const* d_in, const int* in_sizes, int n_in,
                              void* d_out, int out_size, void* d_ws, size_t ws_size,
                              hipStream_t stream) {
    // inputs (setup_inputs order): x_final, x_full, mask, q, head_w, head_b
    const float*         x_full = (const float*)d_in[1];
    const unsigned char* mask   = (const unsigned char*)d_in[2];  // jnp bool = u8
    const float*         q      = (const float*)d_in[3];
    const float*         head_w = (const float*)d_in[4];
    const float*         head_b = (const float*)d_in[5];

    float* out    = (float*)d_out;
    float* scores = out;           // [32]
    float* alpha  = out + BQ;      // [32][4096]

    // workspace layout (floats): ~2.7 MB total
    float* ws        = (float*)d_ws;
    float* logits_ws = ws;                               // B*N       = 131072
    float* part_p    = logits_ws + (size_t)BQ * NQ;      // B*CH*D    = 524288
    float* part_ms   = part_p + (size_t)BQ * CHUNKS * DQ;// B*CH*2    = 1024
    float* pooled_ws = part_ms + (size_t)BQ * CHUNKS * 2;// B*D       = 32768

    probe_pass1<<<dim3(CHUNKS, BQ), 256, 0, stream>>>(x_full, mask, q,
                                                      logits_ws, part_p, part_ms);
    probe_pass2<<<dim3(BQ), 256, 0, stream>>>(part_p, part_ms, logits_ws,
                                              pooled_ws, alpha);
    probe_pass3<<<dim3(1), 512, 0, stream>>>(pooled_ws, head_w, head_b, scores);
}